// STGCNWithHypergraphLearning_39934605918859
// MI455X (gfx1250) — compile-verified
//
#include <hip/hip_runtime.h>

// Problem constants (match reference)
#define Bsz 16
#define Tt  12
#define Nn  1024
#define Dd  64
#define Ee  32
#define MM  (Tt * Nn)        // 12288 tokens per batch
#define TOK (Bsz * MM)       // 196608 tokens total
#define EPSL 1e-5f

typedef __attribute__((ext_vector_type(2))) float v2f;
typedef __attribute__((ext_vector_type(8))) float v8f;

// D(16x16,f32) = A(16x4,f32) * B(4x16,f32) + C
__device__ __forceinline__ v8f wmma4(v2f a, v2f b, v8f c) {
  return __builtin_amdgcn_wmma_f32_16x16x4_f32(
      /*neg_a=*/false, a, /*neg_b=*/false, b,
      /*c_mod=*/(short)0, c, /*reuse_a=*/false, /*reuse_b=*/false);
}

// ---------------------------------------------------------------------------
// Kernel: STGCN branch.  Per 16-token tile:
//   u = 0.5*(x[t-1] + x[t]);  g1 = u@W1^T + b1;  g2 = u@W2^T + b2
//   h = relu(g1*g2) + g1;  out1 = LN(h + x) * gamma + beta
// Block = 128 threads = 4 waves, 1 tile per wave.
// ---------------------------------------------------------------------------
__global__ void __launch_bounds__(128) k_stgcn(
    const float* __restrict__ x, const float* __restrict__ W1,
    const float* __restrict__ b1, const float* __restrict__ W2,
    const float* __restrict__ b2, const float* __restrict__ lg,
    const float* __restrict__ lb, float* __restrict__ out1)
{
  __shared__ float sW1[Dd * Dd];
  __shared__ float sW2[Dd * Dd];
  __shared__ float sU[4][16 * Dd];
  __shared__ float sX[4][16 * Dd];

  const int tid  = threadIdx.x;
  const int wave = tid >> 5;
  const int lane = tid & 31;

  for (int i = tid; i < Dd * Dd; i += 128) { sW1[i] = W1[i]; sW2[i] = W2[i]; }
  __syncthreads();

  const int tile = blockIdx.x * 4 + wave;      // 0..12287, 16 tokens each
  const int m0   = tile * 16;                  // N==1024 => tile stays in one (b,t)
  const int t    = (m0 / Nn) % Tt;

  const float* xr = x + (size_t)m0 * Dd;
  const float* xp = xr - (size_t)Nn * Dd;      // previous frame, same nodes
  for (int idx = lane; idx < 16 * Dd; idx += 32) {
    float cur  = xr[idx];
    float prev = (t > 0) ? xp[idx] : 0.0f;
    sX[wave][idx] = cur;
    sU[wave][idx] = 0.5f * (cur + prev);
  }
  // per-wave LDS tile: DS ops from one wave are in-order; no barrier needed.

  const int lh     = lane & 15;
  const int koff   = (lane >> 4) << 1;         // K sub-offset: 0 or 2
  const int rowoff = (lane >> 4) << 3;         // C row offset: 0 or 8

  v8f acc1[4] = {};
  v8f acc2[4] = {};
  #pragma unroll
  for (int kk = 0; kk < 16; ++kk) {            // K = 64 in steps of 4
    const int kb = kk * 4 + koff;
    v2f a;
    a.x = sU[wave][lh * Dd + kb];
    a.y = sU[wave][lh * Dd + kb + 1];
    #pragma unroll
    for (int ct = 0; ct < 4; ++ct) {           // output cols 0..63 in 16s
      const int o = ct * 16 + lh;
      v2f bb, bc;
      bb.x = sW1[o * Dd + kb];  bb.y = sW1[o * Dd + kb + 1];
      acc1[ct] = wmma4(a, bb, acc1[ct]);
      bc.x = sW2[o * Dd + kb];  bc.y = sW2[o * Dd + kb + 1];
      acc2[ct] = wmma4(a, bc, acc2[ct]);
    }
  }

  float bias1[4], bias2[4], gam[4], bet[4];
  #pragma unroll
  for (int ct = 0; ct < 4; ++ct) {
    const int o = ct * 16 + lh;
    bias1[ct] = b1[o]; bias2[ct] = b2[o]; gam[ct] = lg[o]; bet[ct] = lb[o];
  }

  #pragma unroll
  for (int r = 0; r < 8; ++r) {
    const int rl = r + rowoff;                 // local row 0..15
    float tv[4], s = 0.0f, sq = 0.0f;
    #pragma unroll
    for (int ct = 0; ct < 4; ++ct) {
      float g1 = acc1[ct][r] + bias1[ct];
      float g2 = acc2[ct][r] + bias2[ct];
      float p  = g1 * g2;
      float h  = (p > 0.0f ? p : 0.0f) + g1;
      float v  = h + sX[wave][rl * Dd + ct * 16 + lh];
      tv[ct] = v; s += v; sq += v * v;
    }
    #pragma unroll
    for (int md = 1; md < 16; md <<= 1) {      // 16-lane row reduction
      s  += __shfl_xor(s,  md, 32);
      sq += __shfl_xor(sq, md, 32);
    }
    const float mean = s  * (1.0f / 64.0f);
    const float var  = sq * (1.0f / 64.0f) - mean * mean;
    const float inv  = rsqrtf(var + EPSL);
    float* orow = out1 + (size_t)(m0 + rl) * Dd;
    #pragma unroll
    for (int ct = 0; ct < 4; ++ct)
      orow[ct * 16 + lh] = (tv[ct] - mean) * inv * gam[ct] + bet[ct];
  }
}

// ---------------------------------------------------------------------------
// Kernel: hyper pass 1.  Per 16-token tile (one batch per block group):
//   S = feat @ edge_clf (K=64 -> E=32), assign = softmax_E(S)  -> global+LDS
//   hyper_feat += assign^T(32x16) @ feat(16x64)   (WMMA outer-product accum)
// Grid = B * 48 blocks; block = 128 thr = 4 waves; 4 tiles/wave.
// ---------------------------------------------------------------------------
__global__ void __launch_bounds__(128) k_hyper1(
    const float* __restrict__ x, const float* __restrict__ ec,
    float* __restrict__ assign, float* __restrict__ hf)
{
  __shared__ float sEC[Dd * Ee];               // edge_clf (64x32)
  __shared__ float sFeat[4][16 * Dd];
  __shared__ float sAsn[4][16 * Ee];

  const int tid  = threadIdx.x;
  const int wave = tid >> 5;
  const int lane = tid & 31;
  const int b    = blockIdx.x / 48;
  const int blk  = blockIdx.x % 48;

  for (int i = tid; i < Dd * Ee; i += 128) sEC[i] = ec[i];
  __syncthreads();

  const float* feat = x + (size_t)b * MM * Dd;
  float* asnB       = assign + (size_t)b * MM * Ee;

  const int lh     = lane & 15;
  const int koff   = (lane >> 4) << 1;
  const int rowoff = (lane >> 4) << 3;

  v8f hfa[2][4] = {};                          // E x D accumulator (32x64)

  for (int j = 0; j < 4; ++j) {
    const int tileL = blk * 16 + wave * 4 + j; // 0..767 within batch
    const int m0    = tileL * 16;
    const float* fr = feat + (size_t)m0 * Dd;
    for (int idx = lane; idx < 16 * Dd; idx += 32) sFeat[wave][idx] = fr[idx];

    // S = feat @ edge_clf
    v8f s[2] = {};
    #pragma unroll
    for (int kk = 0; kk < 16; ++kk) {
      const int kb = kk * 4 + koff;
      v2f a;
      a.x = sFeat[wave][lh * Dd + kb];
      a.y = sFeat[wave][lh * Dd + kb + 1];
      #pragma unroll
      for (int ct = 0; ct < 2; ++ct) {
        const int e = ct * 16 + lh;
        v2f bb;
        bb.x = sEC[kb * Ee + e];
        bb.y = sEC[(kb + 1) * Ee + e];
        s[ct] = wmma4(a, bb, s[ct]);
      }
    }

    // row-wise softmax over E=32 (2 values/lane, 16 lanes/row)
    #pragma unroll
    for (int r = 0; r < 8; ++r) {
      const int rl = r + rowoff;
      float v0 = s[0][r], v1 = s[1][r];
      float mx = fmaxf(v0, v1);
      #pragma unroll
      for (int md = 1; md < 16; md <<= 1) mx = fmaxf(mx, __shfl_xor(mx, md, 32));
      float e0 = __expf(v0 - mx), e1 = __expf(v1 - mx);
      float ss = e0 + e1;
      #pragma unroll
      for (int md = 1; md < 16; md <<= 1) ss += __shfl_xor(ss, md, 32);
      const float inv = 1.0f / ss;
      const float a0 = e0 * inv, a1 = e1 * inv;
      float* ar = asnB + (size_t)(m0 + rl) * Ee;
      ar[lh]      = a0;
      ar[16 + lh] = a1;
      sAsn[wave][rl * Ee + lh]      = a0;
      sAsn[wave][rl * Ee + 16 + lh] = a1;
    }

    // hyper_feat accumulation: A = assign^T (E x 16 tokens), B = feat tile
    #pragma unroll
    for (int kk = 0; kk < 4; ++kk) {           // K = 16 tokens
      const int kb = kk * 4 + koff;            // token index
      #pragma unroll
      for (int mt = 0; mt < 2; ++mt) {         // e rows in 16s
        v2f a;
        a.x = sAsn[wave][kb * Ee + mt * 16 + lh];
        a.y = sAsn[wave][(kb + 1) * Ee + mt * 16 + lh];
        #pragma unroll
        for (int nt = 0; nt < 4; ++nt) {       // d cols in 16s
          v2f bb;
          bb.x = sFeat[wave][kb * Dd + nt * 16 + lh];
          bb.y = sFeat[wave][(kb + 1) * Dd + nt * 16 + lh];
          hfa[mt][nt] = wmma4(a, bb, hfa[mt][nt]);
        }
      }
    }
  }

  float* hfB = hf + (size_t)b * Ee * Dd;
  #pragma unroll
  for (int mt = 0; mt < 2; ++mt)
    #pragma unroll
    for (int nt = 0; nt < 4; ++nt)
      #pragma unroll
      for (int r = 0; r < 8; ++r) {
        const int e = mt * 16 + r + rowoff;
        const int d = nt * 16 + lh;
        atomicAdd(&hfB[e * Dd + d], hfa[mt][nt][r]);
      }
}

// ---------------------------------------------------------------------------
// Kernel: hyper pass 2 (tiny): hyper_out = relu(edge_map @ hyper_feat) + hf
// ---------------------------------------------------------------------------
__global__ void k_hyper2(const float* __restrict__ em,
                         const float* __restrict__ hf, float* __restrict__ ho)
{
  const int b = blockIdx.x;
  const float* hfB = hf + (size_t)b * Ee * Dd;
  float* hoB       = ho + (size_t)b * Ee * Dd;
  for (int idx = threadIdx.x; idx < Ee * Dd; idx += blockDim.x) {
    const int e = idx >> 6, d = idx & 63;
    float s = 0.0f;
    #pragma unroll
    for (int f = 0; f < Ee; ++f) s += em[e * Ee + f] * hfB[f * Dd + d];
    hoB[idx] = fmaxf(s, 0.0f) + hfB[idx];
  }
}

// ---------------------------------------------------------------------------
// Kernel: hyper pass 3 + combine:
//   y = relu(assign(16x32) @ hyper_out(32x64));  out2 = LN(y + x)
//   x_next = 0.5 * (out1 + out2)
// ---------------------------------------------------------------------------
__global__ void __launch_bounds__(128) k_combine(
    const float* __restrict__ x, const float* __restrict__ assign,
    const float* __restrict__ ho, const float* __restrict__ out1,
    const float* __restrict__ hg, const float* __restrict__ hb,
    float* __restrict__ xout)
{
  __shared__ float sHO[Ee * Dd];               // 32x64 hyper_out for batch b

  const int tid  = threadIdx.x;
  const int wave = tid >> 5;
  const int lane = tid & 31;
  const int b    = blockIdx.x / 48;
  const int blk  = blockIdx.x % 48;

  for (int i = tid; i < Ee * Dd; i += 128) sHO[i] = ho[(size_t)b * Ee * Dd + i];
  __syncthreads();

  const int lh     = lane & 15;
  const int koff   = (lane >> 4) << 1;
  const int rowoff = (lane >> 4) << 3;

  float gam[4], bet[4];
  #pragma unroll
  for (int nt = 0; nt < 4; ++nt) { gam[nt] = hg[nt * 16 + lh]; bet[nt] = hb[nt * 16 + lh]; }

  for (int j = 0; j < 4; ++j) {
    const int tileL = blk * 16 + wave * 4 + j;
    const size_t gm = (size_t)b * MM + (size_t)tileL * 16;   // global token row base
    const float* ar = assign + gm * Ee;

    v8f y[4] = {};
    #pragma unroll
    for (int kk = 0; kk < 8; ++kk) {           // K = E = 32
      const int kb = kk * 4 + koff;
      v2f a;
      a.x = ar[lh * Ee + kb];
      a.y = ar[lh * Ee + kb + 1];
      #pragma unroll
      for (int nt = 0; nt < 4; ++nt) {
        v2f bb;
        bb.x = sHO[kb * Dd + nt * 16 + lh];
        bb.y = sHO[(kb + 1) * Dd + nt * 16 + lh];
        y[nt] = wmma4(a, bb, y[nt]);
      }
    }

    #pragma unroll
    for (int r = 0; r < 8; ++r) {
      const int rl = r + rowoff;
      const size_t row = gm + rl;
      const float* xr = x + row * Dd;
      float tv[4], s = 0.0f, sq = 0.0f;
      #pragma unroll
      for (int nt = 0; nt < 4; ++nt) {
        float v = fmaxf(y[nt][r], 0.0f) + xr[nt * 16 + lh];
        tv[nt] = v; s += v; sq += v * v;
      }
      #pragma unroll
      for (int md = 1; md < 16; md <<= 1) {
        s  += __shfl_xor(s,  md, 32);
        sq += __shfl_xor(sq, md, 32);
      }
      const float mean = s  * (1.0f / 64.0f);
      const float var  = sq * (1.0f / 64.0f) - mean * mean;
      const float inv  = rsqrtf(var + EPSL);
      const float* o1 = out1 + row * Dd;
      float* xo       = xout + row * Dd;
      #pragma unroll
      for (int nt = 0; nt < 4; ++nt) {
        const float ln = (tv[nt] - mean) * inv * gam[nt] + bet[nt];
        xo[nt * 16 + lh] = 0.5f * (ln + o1[nt * 16 + lh]);
      }
    }
  }
}

__global__ void k_zero(float* __restrict__ p, int n) {
  const int i = blockIdx.x * blockDim.x + threadIdx.x;
  if (i < n) p[i] = 0.0f;
}

// ---------------------------------------------------------------------------
extern "C" void kernel_launch(void* const* d_in, const int* in_sizes, int n_in,
                              void* d_out, int out_size, void* d_ws, size_t ws_size,
                              hipStream_t stream)
{
  (void)in_sizes; (void)n_in; (void)out_size; (void)ws_size;

  const float* x  = (const float*)d_in[0];
  const float* W1 = (const float*)d_in[1];
  const float* b1 = (const float*)d_in[2];
  const float* W2 = (const float*)d_in[3];
  const float* b2 = (const float*)d_in[4];
  const float* lg = (const float*)d_in[5];
  const float* lb = (const float*)d_in[6];
  const float* ec = (const float*)d_in[7];
  const float* em = (const float*)d_in[8];
  const float* hg = (const float*)d_in[9];
  const float* hb = (const float*)d_in[10];

  char* ws = (char*)d_ws;
  float* ws_x  = (float*)ws;  ws += (size_t)TOK * Dd * sizeof(float);   // 50.3 MB
  float* o1buf = (float*)ws;  ws += (size_t)TOK * Dd * sizeof(float);   // 50.3 MB
  float* asn   = (float*)ws;  ws += (size_t)TOK * Ee * sizeof(float);   // 25.2 MB
  float* hfbuf = (float*)ws;  ws += (size_t)Bsz * Ee * Dd * sizeof(float);
  float* hobuf = (float*)ws;

  for (int i = 0; i < 3; ++i) {
    const float* xin = (i == 0) ? x : ws_x;
    float* xo        = (i == 2) ? (float*)d_out : ws_x;   // iter1: safe in-place per-row

    k_zero   <<<(Bsz * Ee * Dd + 255) / 256, 256, 0, stream>>>(hfbuf, Bsz * Ee * Dd);
    k_stgcn  <<<TOK / 64,  128, 0, stream>>>(xin, W1 + i * Dd * Dd, b1 + i * Dd,
                                             W2 + i * Dd * Dd, b2 + i * Dd,
                                             lg + i * Dd, lb + i * Dd, o1buf);
    k_hyper1 <<<Bsz * 48, 128, 0, stream>>>(xin, ec, asn, hfbuf);
    k_hyper2 <<<Bsz, 256, 0, stream>>>(em, hfbuf, hobuf);
    k_combine<<<Bsz * 48, 128, 0, stream>>>(xin, asn, hobuf, o1buf, hg, hb, xo);
  }
}